// SpatialXGModel_65910568124561
// MI455X (gfx1250) — compile-verified
//
#include <hip/hip_runtime.h>
#include <math.h>

typedef __attribute__((ext_vector_type(2))) float v2f;
typedef __attribute__((ext_vector_type(8))) float v8f;

#define TPB 256
#define ENC_NEG_INF 0x007FFFFFu  // order-preserving encoding of -inf

__device__ __forceinline__ unsigned fenc(float f) {
  unsigned u = __float_as_uint(f);
  return (u & 0x80000000u) ? ~u : (u | 0x80000000u);
}
__device__ __forceinline__ float fdec(unsigned u) {
  return (u & 0x80000000u) ? __uint_as_float(u & 0x7fffffffu)
                           : __uint_as_float(~u);
}

// ---------------- init ----------------
__global__ void fill_f32(float* p, float v, long long n) {
  long long i = blockIdx.x * (long long)blockDim.x + threadIdx.x;
  if (i < n) p[i] = v;
}
__global__ void fill_u32(unsigned* p, unsigned v, long long n) {
  long long i = blockIdx.x * (long long)blockDim.x + threadIdx.x;
  if (i < n) p[i] = v;
}

// ---------------- dense ----------------
// h[n,j] = sum_k x[n,k] * W[k,j]   (skinny K=7, VALU path)
__global__ void gemm_small_k(const float* __restrict__ x, const float* __restrict__ W,
                             float* __restrict__ h, int Nn, int K, int F) {
  long long i = blockIdx.x * (long long)blockDim.x + threadIdx.x;
  if (i >= (long long)Nn * F) return;
  int n = (int)(i / F), j = (int)(i % F);
  float acc = 0.f;
  for (int k = 0; k < K; ++k) acc += x[(long long)n * K + k] * W[k * F + j];
  h[i] = acc;
}

// C[M,NC] = A[M,K] @ B[K,NC], fp32 via V_WMMA_F32_16X16X4_F32.
// One wave per 16x16 output tile. K, NC compile-time; K%4==0, NC%16==0.
// EXEC stays full throughout: wave-uniform exit + row clamp (no lane branches).
template <int K, int NC>
__global__ void gemm_wmma_f32(const float* __restrict__ A, const float* __restrict__ B,
                              float* __restrict__ C, int M) {
  int wave = (int)((blockIdx.x * (long long)blockDim.x + threadIdx.x) >> 5);
  int lane = threadIdx.x & 31;
  int tilesM = (M + 15) >> 4;
  int tilesN = NC >> 4;
  if (wave >= tilesM * tilesN) return;          // wave-uniform: EXEC stays full
  int tm = wave / tilesN, tn = wave % tilesN;
  int half = lane >> 4;                          // 0: K pair {0,1}, 1: K pair {2,3}
  int l = lane & 15;
  int arow = tm * 16 + l;
  if (arow >= M) arow = M - 1;                   // clamp: padded A rows feed only
                                                 // padded D rows (stores guarded)
  int ccol = tn * 16 + l;
  const float* Ap = A + (long long)arow * K + 2 * half;  // 8B aligned
  const float* Bp = B + (long long)(2 * half) * NC + ccol;
  v8f acc = {};
#pragma unroll
  for (int k0 = 0; k0 < K; k0 += 4) {
    v2f a = *(const v2f*)(Ap + k0);              // global_load_b64, imm offset
    v2f b;
    b.x = Bp[k0 * NC];                           // imm offsets (K, NC constexpr)
    b.y = Bp[(k0 + 1) * NC];
    acc = __builtin_amdgcn_wmma_f32_16x16x4_f32(false, a, false, b,
                                                (short)0, acc, false, false);
  }
  // D layout: VGPR r -> row tm*16 + half*8 + r, col = tn*16 + (lane&15)
#pragma unroll
  for (int r = 0; r < 8; ++r) {
    int orow = tm * 16 + half * 8 + r;
    if (orow < M) C[(long long)orow * NC + ccol] = acc[r];
  }
}

// ---------------- attention coefficients ----------------
// es[n,h] = <h[n,h,:], a_src[h,:]>,  ed likewise. i indexes (n*H + h).
__global__ void attn_coef(const float* __restrict__ h, const float* __restrict__ a_src,
                          const float* __restrict__ a_dst, float* __restrict__ es,
                          float* __restrict__ ed, int Nn, int H, int C) {
  long long i = blockIdx.x * (long long)blockDim.x + threadIdx.x;
  if (i >= (long long)Nn * H) return;
  int head = (int)(i % H);
  const float* hp = h + i * C;
  float s = 0.f, d = 0.f;
  for (int c = 0; c < C; ++c) {
    float v = hp[c];
    s += v * a_src[head * C + c];
    d += v * a_dst[head * C + c];
  }
  es[i] = s; ed[i] = d;
}

__device__ __forceinline__ void edge_nodes(const int* __restrict__ ei, int e, int E,
                                           int& s, int& d) {
  if (e < E) { s = ei[e]; d = ei[E + e]; }
  else       { s = d = e - E; }                 // appended self-loops
}

// pass 1: e = leaky_relu(es[src]+ed[dst]); store; segment max into menc[dst]
__global__ void edge_max(const int* __restrict__ ei, int E, int ET, int H,
                         const float* __restrict__ es, const float* __restrict__ ed,
                         float* __restrict__ ebuf, unsigned* __restrict__ menc) {
  long long t = blockIdx.x * (long long)blockDim.x + threadIdx.x;
  if (t >= (long long)ET * H) return;
  int e = (int)(t / H), h = (int)(t % H);
  int s, d; edge_nodes(ei, e, E, s, d);
  float v = es[(long long)s * H + h] + ed[(long long)d * H + h];
  v = v > 0.f ? v : 0.2f * v;                    // leaky_relu, PyG default slope
  ebuf[t] = v;
  atomicMax(&menc[(long long)d * H + h], fenc(v));
}

// pass 2: p = exp(e - m[dst]); store; segment sum into den[dst]
__global__ void edge_exp(const int* __restrict__ ei, int E, int ET, int H,
                         const unsigned* __restrict__ menc, float* __restrict__ ebuf,
                         float* __restrict__ den) {
  long long t = blockIdx.x * (long long)blockDim.x + threadIdx.x;
  if (t >= (long long)ET * H) return;
  int e = (int)(t / H), h = (int)(t % H);
  int s, d; edge_nodes(ei, e, E, s, d);
  float p = __expf(ebuf[t] - fdec(menc[(long long)d * H + h]));
  ebuf[t] = p;
  atomicAdd(&den[(long long)d * H + h], p);
}

// pass 3: out[dst,h,:] += h[src,h,:] * (p / (den[dst]+eps)); C%4==0
__global__ void edge_agg(const int* __restrict__ ei, int E, int ET, int H, int C,
                         const float* __restrict__ ebuf, const float* __restrict__ den,
                         const float* __restrict__ hfeat, float* __restrict__ out) {
  long long t = blockIdx.x * (long long)blockDim.x + threadIdx.x;
  if (t >= (long long)ET * H) return;
  int e = (int)(t / H), h = (int)(t % H);
  int s, d; edge_nodes(ei, e, E, s, d);
  float alpha = ebuf[t] / (den[(long long)d * H + h] + 1e-16f);
  const float4* hs4 = (const float4*)(hfeat + ((long long)s * H + h) * C);
  float* od = out + ((long long)d * H + h) * C;
  for (int q = 0; q < (C >> 2); ++q) {
    float4 v = hs4[q];                           // global_load_b128 gather
    atomicAdd(&od[q * 4 + 0], v.x * alpha);
    atomicAdd(&od[q * 4 + 1], v.y * alpha);
    atomicAdd(&od[q * 4 + 2], v.z * alpha);
    atomicAdd(&od[q * 4 + 3], v.w * alpha);
  }
}

// ---------------- activations ----------------
__global__ void bias_elu(float* __restrict__ a, const float* __restrict__ b,
                         long long n, int F) {
  long long i = blockIdx.x * (long long)blockDim.x + threadIdx.x;
  if (i >= n) return;
  float v = a[i] + b[i % F];
  a[i] = v > 0.f ? v : (__expf(v) - 1.f);
}
__global__ void bias_relu(float* __restrict__ a, const float* __restrict__ b,
                          long long n, int F) {
  long long i = blockIdx.x * (long long)blockDim.x + threadIdx.x;
  if (i >= n) return;
  float v = a[i] + b[i % F];
  a[i] = v > 0.f ? v : 0.f;
}

// ---------------- readout ----------------
__global__ void pool(const float* __restrict__ hf, const int* __restrict__ batch,
                     float* __restrict__ cnt, float* __restrict__ sum,
                     unsigned* __restrict__ mx, int Nn, int C) {
  long long i = blockIdx.x * (long long)blockDim.x + threadIdx.x;
  if (i >= Nn) return;
  int b = batch[i];
  atomicAdd(&cnt[b], 1.f);
  for (int c = 0; c < C; ++c) {
    float v = hf[i * (long long)C + c];
    atomicAdd(&sum[(long long)b * C + c], v);
    atomicMax(&mx[(long long)b * C + c], fenc(v));
  }
}
__global__ void build_g(const float* __restrict__ cnt, const float* __restrict__ sum,
                        const unsigned* __restrict__ mx, float* __restrict__ g,
                        int G, int C) {
  long long i = blockIdx.x * (long long)blockDim.x + threadIdx.x;
  if (i >= (long long)G * C) return;
  int gi = (int)(i / C), c = (int)(i % C);
  g[(long long)gi * 2 * C + c]     = sum[i] / cnt[gi];
  g[(long long)gi * 2 * C + C + c] = fdec(mx[i]);
}
__global__ void final_logit(const float* __restrict__ hidden, const float* __restrict__ Wm2,
                            const float* __restrict__ bm2, float* __restrict__ out, int G) {
  int g = blockIdx.x * blockDim.x + threadIdx.x;
  if (g >= G) return;
  float s = bm2[0];
  for (int c = 0; c < 64; ++c) s += hidden[g * 64 + c] * Wm2[c];
  out[g] = s;
}

// ---------------- host ----------------
static inline unsigned gblk(long long n) { return (unsigned)((n + TPB - 1) / TPB); }

extern "C" void kernel_launch(void* const* d_in, const int* in_sizes, int n_in,
                              void* d_out, int out_size, void* d_ws, size_t ws_size,
                              hipStream_t stream) {
  const float* x   = (const float*)d_in[0];
  const int*   ei  = (const int*)  d_in[1];
  const int*   bat = (const int*)  d_in[2];
  const float* W1  = (const float*)d_in[3];
  const float* as1 = (const float*)d_in[4];
  const float* ad1 = (const float*)d_in[5];
  const float* b1  = (const float*)d_in[6];
  const float* W2  = (const float*)d_in[7];
  const float* as2 = (const float*)d_in[8];
  const float* ad2 = (const float*)d_in[9];
  const float* b2  = (const float*)d_in[10];
  const float* Wm1 = (const float*)d_in[11];
  const float* bm1 = (const float*)d_in[12];
  const float* Wm2 = (const float*)d_in[13];
  const float* bm2 = (const float*)d_in[14];
  float* out = (float*)d_out;

  const int Nn = in_sizes[0] / 7;
  const int E  = in_sizes[1] / 2;
  const int ET = E + Nn;
  const int G  = out_size;
  const int H1 = 4, C1 = 32, F1 = 128, HID = 32, FIN = 7;

  // workspace carve-out
  char* w = (char*)d_ws;
  auto carve = [&](size_t bytes) -> void* {
    void* p = (void*)w; w += (bytes + 255) & ~(size_t)255; return p;
  };
  float*    h1    = (float*)   carve((size_t)Nn * F1 * 4);   // layer1 features [N,4,32]
  float*    es1   = (float*)   carve((size_t)Nn * H1 * 4);
  float*    ed1   = (float*)   carve((size_t)Nn * H1 * 4);
  unsigned* m1    = (unsigned*)carve((size_t)Nn * H1 * 4);
  float*    den1  = (float*)   carve((size_t)Nn * H1 * 4);
  float*    p1    = (float*)   carve((size_t)ET * H1 * 4);
  float*    out1  = (float*)   carve((size_t)Nn * F1 * 4);   // -> act1 in place
  float*    h2    = (float*)   carve((size_t)Nn * HID * 4);
  float*    es2   = (float*)   carve((size_t)Nn * 4);
  float*    ed2   = (float*)   carve((size_t)Nn * 4);
  unsigned* m2    = (unsigned*)carve((size_t)Nn * 4);
  float*    den2  = (float*)   carve((size_t)Nn * 4);
  float*    p2    = (float*)   carve((size_t)ET * 4);
  float*    out2  = (float*)   carve((size_t)Nn * HID * 4);  // -> final node feats
  float*    cnt   = (float*)   carve((size_t)G * 4);
  float*    psum  = (float*)   carve((size_t)G * HID * 4);
  unsigned* pmax  = (unsigned*)carve((size_t)G * HID * 4);
  float*    gfeat = (float*)   carve((size_t)G * 2 * HID * 4);
  float*    hid   = (float*)   carve((size_t)G * 64 * 4);
  (void)ws_size; (void)n_in;

  // init
  fill_u32<<<gblk((long long)Nn * H1), TPB, 0, stream>>>(m1, ENC_NEG_INF, (long long)Nn * H1);
  fill_f32<<<gblk((long long)Nn * H1), TPB, 0, stream>>>(den1, 0.f, (long long)Nn * H1);
  fill_f32<<<gblk((long long)Nn * F1), TPB, 0, stream>>>(out1, 0.f, (long long)Nn * F1);
  fill_u32<<<gblk(Nn), TPB, 0, stream>>>(m2, ENC_NEG_INF, Nn);
  fill_f32<<<gblk(Nn), TPB, 0, stream>>>(den2, 0.f, Nn);
  fill_f32<<<gblk((long long)Nn * HID), TPB, 0, stream>>>(out2, 0.f, (long long)Nn * HID);
  fill_f32<<<gblk(G), TPB, 0, stream>>>(cnt, 0.f, G);
  fill_f32<<<gblk((long long)G * HID), TPB, 0, stream>>>(psum, 0.f, (long long)G * HID);
  fill_u32<<<gblk((long long)G * HID), TPB, 0, stream>>>(pmax, ENC_NEG_INF, (long long)G * HID);

  // ---- GAT layer 1 ----
  gemm_small_k<<<gblk((long long)Nn * F1), TPB, 0, stream>>>(x, W1, h1, Nn, FIN, F1);
  attn_coef<<<gblk((long long)Nn * H1), TPB, 0, stream>>>(h1, as1, ad1, es1, ed1, Nn, H1, C1);
  edge_max<<<gblk((long long)ET * H1), TPB, 0, stream>>>(ei, E, ET, H1, es1, ed1, p1, m1);
  edge_exp<<<gblk((long long)ET * H1), TPB, 0, stream>>>(ei, E, ET, H1, m1, p1, den1);
  edge_agg<<<gblk((long long)ET * H1), TPB, 0, stream>>>(ei, E, ET, H1, C1, p1, den1, h1, out1);
  bias_elu<<<gblk((long long)Nn * F1), TPB, 0, stream>>>(out1, b1, (long long)Nn * F1, F1);

  // ---- GAT layer 2 (fp32 WMMA GEMM: [N,128] @ [128,32]) ----
  {
    long long tiles = (long long)((Nn + 15) / 16) * (HID / 16);
    gemm_wmma_f32<128, 32><<<gblk(tiles * 32), TPB, 0, stream>>>(out1, W2, h2, Nn);
  }
  attn_coef<<<gblk(Nn), TPB, 0, stream>>>(h2, as2, ad2, es2, ed2, Nn, 1, HID);
  edge_max<<<gblk(ET), TPB, 0, stream>>>(ei, E, ET, 1, es2, ed2, p2, m2);
  edge_exp<<<gblk(ET), TPB, 0, stream>>>(ei, E, ET, 1, m2, p2, den2);
  edge_agg<<<gblk(ET), TPB, 0, stream>>>(ei, E, ET, 1, HID, p2, den2, h2, out2);
  bias_elu<<<gblk((long long)Nn * HID), TPB, 0, stream>>>(out2, b2, (long long)Nn * HID, HID);

  // ---- readout ----
  pool<<<gblk(Nn), TPB, 0, stream>>>(out2, bat, cnt, psum, pmax, Nn, HID);
  build_g<<<gblk((long long)G * HID), TPB, 0, stream>>>(cnt, psum, pmax, gfeat, G, HID);
  {
    long long tiles = (long long)((G + 15) / 16) * (64 / 16);
    gemm_wmma_f32<64, 64><<<gblk(tiles * 32), TPB, 0, stream>>>(gfeat, Wm1, hid, G);
  }
  bias_relu<<<gblk((long long)G * 64), TPB, 0, stream>>>(hid, bm1, (long long)G * 64, 64);
  final_logit<<<gblk(G), TPB, 0, stream>>>(hid, Wm2, bm2, out, G);
}